// RLGNNLayer_74388833566752
// MI455X (gfx1250) — compile-verified
//
#include <hip/hip_runtime.h>

// GIN-style message passing, 3 layers, fp32, MI455X (gfx1250, wave32).
//   agg = x + scatter_sum(x[src] -> dst)        (bandwidth/atomic bound, ~2.5 GB total)
//   h   = relu(agg @ W1 + b1); x = h @ W2 + b2  (fp32 WMMA 16x16x4)
// MLP kernel: A strip in registers, weight tiles staged per-workgroup in LDS,
// so the K-loops are pure v_wmma + ds_load (no global loads in steady state).

#define NN 50000
#define NE 800000
#define D  128   // D_IN == D_OUT
#define DH 256   // hidden
#define NTILES (NN / 16)   // 3125 sixteen-row tiles (NN % 16 == 0)

typedef __attribute__((ext_vector_type(2))) float v2f;
typedef __attribute__((ext_vector_type(8))) float v8f;

// ---------------------------------------------------------------- agg = x ---
__global__ void gin_init_agg(float4* __restrict__ agg, const float4* __restrict__ xin, int n4) {
  int i = blockIdx.x * blockDim.x + threadIdx.x;
  if (i < n4) agg[i] = xin[i];
}

// ------------------------------------------- scatter: one wave per edge ----
__global__ __launch_bounds__(256) void gin_scatter(const float* __restrict__ xin,
                                                   const int* __restrict__ src,
                                                   const int* __restrict__ dst,
                                                   float* __restrict__ agg) {
  int e    = blockIdx.x * (blockDim.x >> 5) + (threadIdx.x >> 5);
  int lane = threadIdx.x & 31;
  if (e >= NE) return;
  int s = src[e];
  int d = dst[e];
  const float4 v = *(const float4*)(xin + (size_t)s * D + lane * 4);
  float* base = agg + (size_t)d * D + lane * 4;
  atomicAdd(base + 0, v.x);
  atomicAdd(base + 1, v.y);
  atomicAdd(base + 2, v.z);
  atomicAdd(base + 3, v.w);
}

// ----------------------- fused 2-layer MLP: one wave per 16-row node slab ---
__global__ __launch_bounds__(256) void gin_mlp(const float* __restrict__ xa,   // [N,128] = x+agg
                                               const float* __restrict__ W1l,  // [128,256]
                                               const float* __restrict__ b1l,  // [256]
                                               const float* __restrict__ W2l,  // [256,128]
                                               const float* __restrict__ b2l,  // [128]
                                               float* __restrict__ xout) {     // [N,128]
  __shared__ float lw1[128 * 16];   // W1 column-tile, [k][j]       (8 KB, conflict-free B reads)
  __shared__ float lw2[16 * 144];   // W2 row-tile,   [kk][n] p144  (9 KB, halves on disjoint banks)
  __shared__ float lh[8 * 272];     // per-wave h tile [r][c] p17   (8.5 KB)

  const int tid  = threadIdx.x;
  const int wv   = tid >> 5;
  const int lane = tid & 31;
  const int nlo  = lane & 15;                 // N index (B/C/D) or M row (A)
  const int hi   = lane >> 4;                 // which K pair of the 4-wide step
  int tile = blockIdx.x * 8 + wv;
  if (tile >= NTILES) tile = NTILES - 1;      // clamp, keep every wave in the barriers
  const int r0 = tile * 16;

  float* hw = lh + wv * 272;

  // ---- preload this lane's A strip into registers: row r0+nlo, cols {4t+2hi,+1} ----
  const float* arow = xa + (size_t)(r0 + nlo) * D + 2 * hi;
  v2f areg[32];
#pragma unroll
  for (int t = 0; t < 32; ++t) areg[t] = *(const v2f*)(arow + 4 * t);

  v8f zero = {};
  v8f oacc[8];                                // 16x128 output accumulator
#pragma unroll
  for (int mt = 0; mt < 8; ++mt) oacc[mt] = zero;

  for (int nt = 0; nt < 16; ++nt) {           // 16 column-tiles of h
    // ---- cooperative stage of this nt's weight tiles (2048 floats each) ----
#pragma unroll
    for (int i = 0; i < 8; ++i) {
      const int idx = tid + i * 256;
      lw1[idx] = W1l[(idx >> 4) * DH + nt * 16 + (idx & 15)];
      lw2[(idx >> 7) * 144 + (idx & 127)] = W2l[(nt * 16 + (idx >> 7)) * D + (idx & 127)];
    }
    __syncthreads();

    // ---- GEMM1: h_tile[16x16] = xa[16x128] @ W1[:, nt*16..] ----------------
    v8f acc = zero;
#pragma unroll
    for (int k0 = 0; k0 < D; k0 += 4) {
      const int ka = k0 + hi * 2;
      v2f b;
      b.x = lw1[(ka + 0) * 16 + nlo];
      b.y = lw1[(ka + 1) * 16 + nlo];
      acc = __builtin_amdgcn_wmma_f32_16x16x4_f32(false, areg[k0 >> 2], false, b,
                                                  (short)0, acc, false, false);
    }

    // ---- bias + ReLU, park tile in this wave's LDS region ------------------
    const float b1v = b1l[nt * 16 + nlo];
#pragma unroll
    for (int i = 0; i < 8; ++i) {
      float h = acc[i] + b1v;                 // C layout: row = i + 8*hi, col = nlo
      hw[(i + 8 * hi) * 17 + nlo] = h > 0.f ? h : 0.f;
    }
    __asm__ volatile("s_wait_dscnt 0x0" ::: "memory");  // cross-lane LDS RAW, same wave

    // ---- GEMM2 partial: oacc += h_tile[16x16] @ W2[nt*16.., :] -------------
#pragma unroll
    for (int mt = 0; mt < 8; ++mt) {
#pragma unroll
      for (int k0 = 0; k0 < 16; k0 += 4) {
        const int ka = k0 + hi * 2;
        v2f a, b;
        a.x = hw[nlo * 17 + ka];              // A layout: row = nlo, cols ka, ka+1
        a.y = hw[nlo * 17 + ka + 1];
        b.x = lw2[(ka + 0) * 144 + mt * 16 + nlo];
        b.y = lw2[(ka + 1) * 144 + mt * 16 + nlo];
        oacc[mt] = __builtin_amdgcn_wmma_f32_16x16x4_f32(false, a, false, b, (short)0,
                                                         oacc[mt], false, false);
      }
    }
    __syncthreads();                          // protect lw1/lw2 before next stage
  }

  // ---- bias2 + store --------------------------------------------------------
#pragma unroll
  for (int mt = 0; mt < 8; ++mt) {
    const float b2v = b2l[mt * 16 + nlo];
#pragma unroll
    for (int i = 0; i < 8; ++i) {
      xout[(size_t)(r0 + i + 8 * hi) * D + mt * 16 + nlo] = oacc[mt][i] + b2v;
    }
  }
}

// ---------------------------------------------------------------------------
extern "C" void kernel_launch(void* const* d_in, const int* in_sizes, int n_in,
                              void* d_out, int out_size, void* d_ws, size_t ws_size,
                              hipStream_t stream) {
  const float* x0 = (const float*)d_in[0];
  const float* W1 = (const float*)d_in[1];
  const float* b1 = (const float*)d_in[2];
  const float* W2 = (const float*)d_in[3];
  const float* b2 = (const float*)d_in[4];
  const int* src  = (const int*)d_in[5];
  const int* dst  = (const int*)d_in[6];
  float* agg = (float*)d_ws;        // [N,128] scratch (25.6 MB)
  float* out = (float*)d_out;       // doubles as inter-layer x buffer (exact size)

  const int n4 = NN * D / 4;
  for (int l = 0; l < 3; ++l) {
    const float* xin = (l == 0) ? x0 : out;
    gin_init_agg<<<(n4 + 255) / 256, 256, 0, stream>>>((float4*)agg, (const float4*)xin, n4);
    gin_scatter<<<(NE + 7) / 8, 256, 0, stream>>>(xin, src, dst, agg);
    gin_mlp<<<(NTILES + 7) / 8, 256, 0, stream>>>(agg, W1 + l * D * DH, b1 + l * DH,
                                                  W2 + l * DH * D, b2 + l * D, out);
  }
}